// PredictiveCodingNetwork_22497038696513
// MI455X (gfx1250) — compile-verified
//
#include <hip/hip_runtime.h>
#include <hip/hip_bf16.h>
#include <math.h>

// Problem constants (match reference)
#define NLAYERS 4
#define NSTEPS  8
#define DDIM    1024
#define BBATCH  8
#define SSEQ    512
#define NTOK    (BBATCH * SSEQ)        // 4096 tokens
#define VVOCAB  32000
#define CETOK   (BBATCH * (SSEQ - 1))  // 4088 CE tokens
#define LRS     0.1f

typedef float v2f __attribute__((ext_vector_type(2)));
typedef float v8f __attribute__((ext_vector_type(8)));
typedef int   i32x4 __attribute__((vector_size(16)));

// Exact-fp32 CDNA5 WMMA: D(16x16 f32) = A(16x4 f32) * B(4x16 f32) + C
__device__ __forceinline__ v8f wmma4(v2f a, v2f b, v8f c) {
    return __builtin_amdgcn_wmma_f32_16x16x4_f32(
        /*neg_a=*/false, a, /*neg_b=*/false, b,
        /*c_mod=*/(short)0, c, /*reuse_a=*/false, /*reuse_b=*/false);
}

// ---- CDNA5 async global->LDS copies (ASYNCcnt), with portable fallback ----
#if defined(__gfx1250__) && __has_builtin(__builtin_amdgcn_global_load_async_to_lds_b128)
#define HAS_ASYNC_LDS 1
#else
#define HAS_ASYNC_LDS 0
#endif

#define AS1 __attribute__((address_space(1)))
#define AS3 __attribute__((address_space(3)))

__device__ __forceinline__ void cp16(void* lds, const void* g) {
#if HAS_ASYNC_LDS
    __builtin_amdgcn_global_load_async_to_lds_b128(
        (AS1 i32x4*)g, (AS3 i32x4*)lds, 0, 0);
#else
    *(float4*)lds = *(const float4*)g;
#endif
}

__device__ __forceinline__ void cp4(void* lds, const void* g) {
#if HAS_ASYNC_LDS && __has_builtin(__builtin_amdgcn_global_load_async_to_lds_b32)
    __builtin_amdgcn_global_load_async_to_lds_b32(
        (AS1 int*)g, (AS3 int*)lds, 0, 0);
#else
    *(float*)lds = *(const float*)g;
#endif
}

__device__ __forceinline__ void async_wait_all() {
#if HAS_ASYNC_LDS
#if __has_builtin(__builtin_amdgcn_s_wait_asynccnt)
    __builtin_amdgcn_s_wait_asynccnt(0);
#else
    asm volatile("s_wait_asynccnt 0x0" ::: "memory");
#endif
#endif
}

// ---------------------------------------------------------------------------
// Embedding gather: x[t,:] = emb[ids[t],:]
// ---------------------------------------------------------------------------
__global__ void gather_kernel(const int* __restrict__ ids,
                              const float* __restrict__ emb,
                              float* __restrict__ x) {
    const int t  = blockIdx.x;
    const int id = ids[t];
    const float4* src = (const float4*)(emb + (size_t)id * DDIM);
    float4*       dst = (float4*)(x + (size_t)t * DDIM);
    for (int i = threadIdx.x; i < DDIM / 4; i += blockDim.x) dst[i] = src[i];
}

__global__ void zero2_kernel(float* o) {
    if (threadIdx.x < 2) o[threadIdx.x] = 0.0f;
}

// ---------------------------------------------------------------------------
// eps kernel:  E_l = mu[l] - (mu[l+1] @ W_l^T + b_l)
//   mu[0] = x, mu[i] = states[i-1].  A = states[l], B_eff[k,n] = W[n,k].
// 128x128 tile per WG; 8 waves, wave = 32x64 (2x4 WMMA tiles).
// Double-buffered async staging; B kept transposed Bt[n][k] -> b64 frag loads.
// ---------------------------------------------------------------------------
__global__ __launch_bounds__(256)
void eps_kernel(const float* __restrict__ states, const float* __restrict__ x,
                const float* __restrict__ Wst, const float* __restrict__ bst,
                float* __restrict__ E) {
    const int l = blockIdx.z;
    const float* A     = states + (size_t)l * NTOK * DDIM;     // mu[l+1]
    const float* W     = Wst + (size_t)l * DDIM * DDIM;
    const float* bias  = bst + l * DDIM;
    const float* lower = (l == 0) ? x : states + (size_t)(l - 1) * NTOK * DDIM;
    float*       out   = E + (size_t)l * NTOK * DDIM;

    // stride 20 floats: 16B-aligned rows, (20*m+k)%64 distinct for m=0..15
    __shared__ float As[2][128][20];
    __shared__ float Bt[2][128][20];

    const int tid  = threadIdx.x;
    const int lane = tid & 31;
    const int wave = tid >> 5;
    const int m0 = blockIdx.y * 128;
    const int n0 = blockIdx.x * 128;
    const int wm = (wave >> 1) * 32;
    const int wn = (wave & 1) * 64;
    const int lm  = lane & 15;
    const int lk  = (lane >> 4) * 2;
    const int lhi = (lane >> 4) * 8;

    // fixed staging assignment: 2 x 16B segments per tile per thread
    const int srow0 = tid >> 2,         scol0 = (tid & 3) * 4;
    const int srow1 = (tid + 256) >> 2, scol1 = ((tid + 256) & 3) * 4;

    v8f z = {};
    v8f acc[2][4];
    #pragma unroll
    for (int i = 0; i < 2; ++i)
        #pragma unroll
        for (int j = 0; j < 4; ++j) acc[i][j] = z;

    // prologue: stage chunk 0
    {
        const int k0 = 0;
        cp16(&As[0][srow0][scol0], A + (size_t)(m0 + srow0) * DDIM + k0 + scol0);
        cp16(&As[0][srow1][scol1], A + (size_t)(m0 + srow1) * DDIM + k0 + scol1);
        cp16(&Bt[0][srow0][scol0], W + (size_t)(n0 + srow0) * DDIM + k0 + scol0);
        cp16(&Bt[0][srow1][scol1], W + (size_t)(n0 + srow1) * DDIM + k0 + scol1);
    }
    async_wait_all();
    __syncthreads();

    const int nch = DDIM / 16;
    for (int c = 0; c < nch; ++c) {
        const int cur = c & 1;
        if (c + 1 < nch) {
            const int k0 = (c + 1) * 16, nb = cur ^ 1;
            cp16(&As[nb][srow0][scol0], A + (size_t)(m0 + srow0) * DDIM + k0 + scol0);
            cp16(&As[nb][srow1][scol1], A + (size_t)(m0 + srow1) * DDIM + k0 + scol1);
            cp16(&Bt[nb][srow0][scol0], W + (size_t)(n0 + srow0) * DDIM + k0 + scol0);
            cp16(&Bt[nb][srow1][scol1], W + (size_t)(n0 + srow1) * DDIM + k0 + scol1);
        }
        #pragma unroll
        for (int kk = 0; kk < 4; ++kk) {
            const int kb = kk * 4 + lk;
            v2f a[2], b[4];
            #pragma unroll
            for (int i = 0; i < 2; ++i) {
                const float* p = &As[cur][wm + i * 16 + lm][kb];
                a[i].x = p[0]; a[i].y = p[1];
            }
            #pragma unroll
            for (int j = 0; j < 4; ++j) {
                const float* p = &Bt[cur][wn + j * 16 + lm][kb];
                b[j].x = p[0]; b[j].y = p[1];
            }
            #pragma unroll
            for (int i = 0; i < 2; ++i)
                #pragma unroll
                for (int j = 0; j < 4; ++j)
                    acc[i][j] = wmma4(a[i], b[j], acc[i][j]);
        }
        async_wait_all();
        __syncthreads();
    }

    #pragma unroll
    for (int i = 0; i < 2; ++i)
        #pragma unroll
        for (int j = 0; j < 4; ++j)
            #pragma unroll
            for (int v = 0; v < 8; ++v) {
                int row = m0 + wm + i * 16 + lhi + v;
                int col = n0 + wn + j * 16 + lm;
                size_t idx = (size_t)row * DDIM + col;
                out[idx] = lower[idx] - (acc[i][j][v] + bias[col]);
            }
}

// ---------------------------------------------------------------------------
// update kernel: states[l] += 0.1*(E_l @ W_l) - 0.1*E_{l+1} (last layer: none)
// B = W untransposed in memory -> transposed LDS tile via async b32 gather
// (global-coalesced along n), identical b64-fragment inner loop.
// ---------------------------------------------------------------------------
__global__ __launch_bounds__(256)
void update_kernel(float* __restrict__ states, const float* __restrict__ Wst,
                   const float* __restrict__ E) {
    const int l = blockIdx.z;
    const float* A = E + (size_t)l * NTOK * DDIM;              // E_l
    const float* W = Wst + (size_t)l * DDIM * DDIM;
    float*       S = states + (size_t)l * NTOK * DDIM;
    const float* Eself = (l < NLAYERS - 1) ? (E + (size_t)(l + 1) * NTOK * DDIM)
                                           : nullptr;

    __shared__ float As[2][128][20];
    __shared__ float Bt[2][128][20];

    const int tid  = threadIdx.x;
    const int lane = tid & 31;
    const int wave = tid >> 5;
    const int m0 = blockIdx.y * 128;
    const int n0 = blockIdx.x * 128;
    const int wm = (wave >> 1) * 32;
    const int wn = (wave & 1) * 64;
    const int lm  = lane & 15;
    const int lk  = (lane >> 4) * 2;
    const int lhi = (lane >> 4) * 8;

    const int srow0 = tid >> 2,         scol0 = (tid & 3) * 4;
    const int srow1 = (tid + 256) >> 2, scol1 = ((tid + 256) & 3) * 4;

    v8f z = {};
    v8f acc[2][4];
    #pragma unroll
    for (int i = 0; i < 2; ++i)
        #pragma unroll
        for (int j = 0; j < 4; ++j) acc[i][j] = z;

    {
        const int k0 = 0;
        cp16(&As[0][srow0][scol0], A + (size_t)(m0 + srow0) * DDIM + k0 + scol0);
        cp16(&As[0][srow1][scol1], A + (size_t)(m0 + srow1) * DDIM + k0 + scol1);
        #pragma unroll
        for (int j = 0; j < 8; ++j) {          // Bt[n][k] = W[(k0+k)*D + n0+n]
            int e = tid + 256 * j, n = e & 127, k = e >> 7;
            cp4(&Bt[0][n][k], W + (size_t)(k0 + k) * DDIM + n0 + n);
        }
    }
    async_wait_all();
    __syncthreads();

    const int nch = DDIM / 16;
    for (int c = 0; c < nch; ++c) {
        const int cur = c & 1;
        if (c + 1 < nch) {
            const int k0 = (c + 1) * 16, nb = cur ^ 1;
            cp16(&As[nb][srow0][scol0], A + (size_t)(m0 + srow0) * DDIM + k0 + scol0);
            cp16(&As[nb][srow1][scol1], A + (size_t)(m0 + srow1) * DDIM + k0 + scol1);
            #pragma unroll
            for (int j = 0; j < 8; ++j) {
                int e = tid + 256 * j, n = e & 127, k = e >> 7;
                cp4(&Bt[nb][n][k], W + (size_t)(k0 + k) * DDIM + n0 + n);
            }
        }
        #pragma unroll
        for (int kk = 0; kk < 4; ++kk) {
            const int kb = kk * 4 + lk;
            v2f a[2], b[4];
            #pragma unroll
            for (int i = 0; i < 2; ++i) {
                const float* p = &As[cur][wm + i * 16 + lm][kb];
                a[i].x = p[0]; a[i].y = p[1];
            }
            #pragma unroll
            for (int j = 0; j < 4; ++j) {
                const float* p = &Bt[cur][wn + j * 16 + lm][kb];
                b[j].x = p[0]; b[j].y = p[1];
            }
            #pragma unroll
            for (int i = 0; i < 2; ++i)
                #pragma unroll
                for (int j = 0; j < 4; ++j)
                    acc[i][j] = wmma4(a[i], b[j], acc[i][j]);
        }
        async_wait_all();
        __syncthreads();
    }

    #pragma unroll
    for (int i = 0; i < 2; ++i)
        #pragma unroll
        for (int j = 0; j < 4; ++j)
            #pragma unroll
            for (int v = 0; v < 8; ++v) {
                int row = m0 + wm + i * 16 + lhi + v;
                int col = n0 + wn + j * 16 + lm;
                size_t idx = (size_t)row * DDIM + col;
                float val = S[idx] + LRS * acc[i][j][v];
                if (Eself) val -= LRS * Eself[idx];
                S[idx] = val;
            }
}

// ---------------------------------------------------------------------------
// pc_loss: sum(E^2) / (L*NTOK*D) accumulated into out[1]
// ---------------------------------------------------------------------------
__global__ __launch_bounds__(256)
void pcloss_kernel(const float* __restrict__ E, float* __restrict__ out) {
    const size_t n = (size_t)NLAYERS * NTOK * DDIM;
    float s = 0.0f;
    for (size_t i = (size_t)blockIdx.x * 256 + threadIdx.x; i < n;
         i += (size_t)gridDim.x * 256) {
        float v = E[i];
        s += v * v;
    }
    for (int off = 16; off > 0; off >>= 1) s += __shfl_xor(s, off);
    __shared__ float wsum[8];
    if ((threadIdx.x & 31) == 0) wsum[threadIdx.x >> 5] = s;
    __syncthreads();
    if (threadIdx.x < 8) {
        float t = wsum[threadIdx.x];
        t += __shfl_xor(t, 4);
        t += __shfl_xor(t, 2);
        t += __shfl_xor(t, 1);
        if (threadIdx.x == 0)
            atomicAdd(out + 1, t * (1.0f / ((float)NLAYERS * NTOK * DDIM)));
    }
}

// ---------------------------------------------------------------------------
// CE kernel: streaming logits = h @ out_W^T + out_b with online log-softmax.
// 64 token rows per WG; vocab in chunks of 128 (16 per wave); K staged by 64
// via async copies. Logits never touch memory.
// ---------------------------------------------------------------------------
__global__ __launch_bounds__(256)
void ce_kernel(const float* __restrict__ h, const float* __restrict__ oW,
               const float* __restrict__ ob, const int* __restrict__ targets,
               float* __restrict__ out) {
    // stride 68 floats: 16B-aligned rows, (4r+k)%64 distinct for r=0..15
    __shared__ float hs[64][68];
    __shared__ float Wv[128][68];
    __shared__ float statM[8][64];
    __shared__ float statS[8][64];
    __shared__ float tgtl[64];
    __shared__ int   rowid[64];
    __shared__ int   tgts[64];

    const int tid  = threadIdx.x;
    const int lane = tid & 31;
    const int wave = tid >> 5;
    const int lm   = lane & 15;
    const int lk   = (lane >> 4) * 2;
    const int lhi  = (lane >> 4) * 8;
    const int t0   = blockIdx.x * 64;

    if (tid < 64) {
        int t  = t0 + tid;
        int tc = (t < CETOK) ? t : (CETOK - 1);
        int b  = tc / (SSEQ - 1);
        int s  = tc % (SSEQ - 1);
        rowid[tid] = b * SSEQ + s;
        tgts[tid]  = targets[b * SSEQ + s + 1];
        tgtl[tid]  = 0.0f;
    }
    __syncthreads();

    // fixed staging assignment: hs gets 4 segments / thread, Wv gets 8
    int hrow[4];
    #pragma unroll
    for (int j = 0; j < 4; ++j) hrow[j] = rowid[(tid + 256 * j) >> 4];

    float m_run[4][8], s_run[4][8];
    #pragma unroll
    for (int t = 0; t < 4; ++t)
        #pragma unroll
        for (int v = 0; v < 8; ++v) { m_run[t][v] = -3.0e38f; s_run[t][v] = 0.0f; }

    for (int vc = 0; vc < VVOCAB / 128; ++vc) {
        const int vbase = vc * 128;
        const int v0w   = vbase + wave * 16;
        const float obv = ob[v0w + lm];

        v8f z = {};
        v8f acc[4];
        #pragma unroll
        for (int t = 0; t < 4; ++t) acc[t] = z;

        for (int kc = 0; kc < DDIM; kc += 64) {
            __syncthreads();
            #pragma unroll
            for (int j = 0; j < 4; ++j) {
                int seg = tid + 256 * j, m = seg >> 4, sc = (seg & 15) * 4;
                cp16(&hs[m][sc], h + (size_t)hrow[j] * DDIM + kc + sc);
            }
            #pragma unroll
            for (int j = 0; j < 8; ++j) {
                int seg = tid + 256 * j, r = seg >> 4, sc = (seg & 15) * 4;
                cp16(&Wv[r][sc], oW + (size_t)(vbase + r) * DDIM + kc + sc);
            }
            async_wait_all();
            __syncthreads();
            #pragma unroll
            for (int kk = 0; kk < 16; ++kk) {
                const int kb = kk * 4 + lk;
                v2f bf;
                const float* pb = &Wv[wave * 16 + lm][kb];
                bf.x = pb[0]; bf.y = pb[1];
                #pragma unroll
                for (int t = 0; t < 4; ++t) {
                    v2f a;
                    const float* pa = &hs[t * 16 + lm][kb];
                    a.x = pa[0]; a.y = pa[1];
                    acc[t] = wmma4(a, bf, acc[t]);
                }
            }
        }

        // online softmax update; 16-lane half-wave owns one row's 16 columns
        const int c = v0w + lm;
        #pragma unroll
        for (int t = 0; t < 4; ++t) {
            #pragma unroll
            for (int v = 0; v < 8; ++v) {
                float val = acc[t][v] + obv;
                int rl = t * 16 + lhi + v;
                if (t0 + rl < CETOK && tgts[rl] == c) tgtl[rl] = val;
                float tmax = val;
                tmax = fmaxf(tmax, __shfl_xor(tmax, 1));
                tmax = fmaxf(tmax, __shfl_xor(tmax, 2));
                tmax = fmaxf(tmax, __shfl_xor(tmax, 4));
                tmax = fmaxf(tmax, __shfl_xor(tmax, 8));
                float mnew = fmaxf(m_run[t][v], tmax);
                float e = __expf(val - mnew);
                e += __shfl_xor(e, 1);
                e += __shfl_xor(e, 2);
                e += __shfl_xor(e, 4);
                e += __shfl_xor(e, 8);
                s_run[t][v] = s_run[t][v] * __expf(m_run[t][v] - mnew) + e;
                m_run[t][v] = mnew;
            }
        }
    }

    if (lm == 0) {  // lanes 0 and 16 hold stats for rows lhi..lhi+7 of each tile
        #pragma unroll
        for (int t = 0; t < 4; ++t)
            #pragma unroll
            for (int v = 0; v < 8; ++v) {
                int rl = t * 16 + lhi + v;
                statM[wave][rl] = m_run[t][v];
                statS[wave][rl] = s_run[t][v];
            }
    }
    __syncthreads();
    if (tid < 64 && (t0 + tid) < CETOK) {
        float m = -3.0e38f;
        #pragma unroll
        for (int w = 0; w < 8; ++w) m = fmaxf(m, statM[w][tid]);
        float s = 0.0f;
        #pragma unroll
        for (int w = 0; w < 8; ++w) s += statS[w][tid] * __expf(statM[w][tid] - m);
        float lse = m + __logf(s);
        atomicAdd(out, (lse - tgtl[tid]) * (1.0f / (float)CETOK));
    }
}

// ---------------------------------------------------------------------------
extern "C" void kernel_launch(void* const* d_in, const int* in_sizes, int n_in,
                              void* d_out, int out_size, void* d_ws, size_t ws_size,
                              hipStream_t stream) {
    const int*   input_ids = (const int*)d_in[0];
    const int*   targets   = (const int*)d_in[1];
    const float* emb       = (const float*)d_in[2];
    const float* W_stack   = (const float*)d_in[3];
    const float* b_stack   = (const float*)d_in[4];
    const float* out_W     = (const float*)d_in[5];
    const float* out_b     = (const float*)d_in[6];
    const float* states0   = (const float*)d_in[7];
    float* out = (float*)d_out;

    float* ws = (float*)d_ws;
    const size_t plane = (size_t)NTOK * DDIM;   // 4M floats
    float* x      = ws;                         // 1 plane  (16 MB)
    float* states = ws + plane;                 // 4 planes (64 MB)
    float* E      = ws + plane * 5;             // 4 planes (64 MB)

    (void)hipMemcpyAsync(states, states0, (size_t)NLAYERS * plane * sizeof(float),
                         hipMemcpyDeviceToDevice, stream);
    gather_kernel<<<NTOK, 256, 0, stream>>>(input_ids, emb, x);
    zero2_kernel<<<1, 32, 0, stream>>>(out);

    dim3 grid(DDIM / 128, NTOK / 128, NLAYERS);
    dim3 blk(256);
    for (int step = 0; step < NSTEPS; ++step) {
        eps_kernel<<<grid, blk, 0, stream>>>(states, x, W_stack, b_stack, E);
        update_kernel<<<grid, blk, 0, stream>>>(states, W_stack, E);
    }
    // final eps pass for pc_loss (post-scan mu)
    eps_kernel<<<grid, blk, 0, stream>>>(states, x, W_stack, b_stack, E);
    pcloss_kernel<<<1024, 256, 0, stream>>>(E, out);

    ce_kernel<<<(CETOK + 63) / 64, 256, 0, stream>>>(
        states + (size_t)(NLAYERS - 1) * plane, out_W, out_b, targets, out);
}